// GlobalTemporalCL_38019050504455
// MI455X (gfx1250) — compile-verified
//
#include <hip/hip_runtime.h>
#include <math.h>

typedef __attribute__((ext_vector_type(2))) float v2f;
typedef __attribute__((ext_vector_type(8))) float v8f;

#define B_  4096
#define S_  512
#define C_  64
#define T_  8

// ---------------------------------------------------------------------------
// Kernel 1: IM_Cx[b,c] = relu( (1/(t+1)) * sum_{s<=t} Main[b,s,:] @ W_p^T + b_p )
// Streams 269 MB of Main_x with b128 loads (full 256B rows per wave).
// ---------------------------------------------------------------------------
__global__ void k_imcx(const float* __restrict__ Main, const int* __restrict__ tp,
                       const float* __restrict__ Wp, const float* __restrict__ bp,
                       float* __restrict__ IM) {
  const int b = blockIdx.x;
  const int tid = threadIdx.x;
  const int g  = tid >> 4;            // 16 s-groups
  const int cq = (tid & 15) * 4;      // float4 column
  const int len = *tp + 1;
  const float* base = Main + (size_t)b * S_ * C_;
  float4 a4 = {0.f, 0.f, 0.f, 0.f};
  for (int s = g; s < len; s += 16) {
    float4 v = *(const float4*)(base + (size_t)s * C_ + cq);
    a4.x += v.x; a4.y += v.y; a4.z += v.z; a4.w += v.w;
  }
  __shared__ __align__(16) float red[16][64];
  __shared__ float meanv[64];
  *(float4*)&red[g][cq] = a4;
  __syncthreads();
  if (tid < 64) {
    float s = 0.f;
#pragma unroll
    for (int q = 0; q < 16; ++q) s += red[q][tid];
    meanv[tid] = s / (float)len;
  }
  __syncthreads();
  const int c = tid & 63, h0 = (tid >> 6) * 16;
  float p = 0.f;
#pragma unroll
  for (int h = 0; h < 16; ++h) p += meanv[h0 + h] * Wp[c * 64 + h0 + h];
  red[tid >> 6][c] = p;
  __syncthreads();
  if (tid < 64) {
    float v = red[0][tid] + red[1][tid] + red[2][tid] + red[3][tid] + bp[tid];
    IM[(size_t)b * 64 + tid] = fmaxf(v, 0.f);
  }
}

// ---------------------------------------------------------------------------
// Kernel 2: Pred[t,i,c] = sum_h IM[i,h] * W_map[t,c,h] + b_map[t,c]   (WMMA f32)
// ---------------------------------------------------------------------------
__global__ void k_pred(const float* __restrict__ IM, const float* __restrict__ Wmap,
                       const float* __restrict__ bmap, float* __restrict__ Pred) {
  const int t  = blockIdx.x >> 7;
  const int rb = (blockIdx.x & 127) * 32;
  const int tid = threadIdx.x;
  __shared__ __align__(16) float sA[32][68];
  __shared__ __align__(16) float sB[64][68];
  {
    int r = tid >> 3;
    int cq = (tid & 7) * 8;
    const float* src = IM + (size_t)(rb + r) * 64 + cq;
    *(float4*)&sA[r][cq]     = *(const float4*)src;
    *(float4*)&sA[r][cq + 4] = *(const float4*)(src + 4);
  }
  {
    int r = tid >> 2;
    int cq = (tid & 3) * 16;
    const float* src = Wmap + (size_t)t * 4096 + (size_t)r * 64 + cq;
#pragma unroll
    for (int q = 0; q < 4; ++q)
      *(float4*)&sB[r][cq + q * 4] = *(const float4*)(src + q * 4);
  }
  __syncthreads();
  const int wave = tid >> 5, lane = tid & 31;
  const int colT = wave & 3, rowT = wave >> 2;
  const int n = lane & 15, kb = (lane >> 4) * 2;
  const float* aRow = &sA[rowT * 16 + n][0];
  const float* bRow = &sB[colT * 16 + n][0];
  v8f acc = {0.f, 0.f, 0.f, 0.f, 0.f, 0.f, 0.f, 0.f};
#pragma unroll
  for (int k0 = 0; k0 < 64; k0 += 4) {
    v2f a = *(const v2f*)(aRow + k0 + kb);
    v2f b = *(const v2f*)(bRow + k0 + kb);
    acc = __builtin_amdgcn_wmma_f32_16x16x4_f32(false, a, false, b, (short)0, acc,
                                                false, false);
  }
  const int cidx = colT * 16 + n;
  const float bm = bmap[t * 64 + cidx];
  const int rbase = rb + rowT * 16 + (lane >> 4) * 8;
#pragma unroll
  for (int j = 0; j < 8; ++j)
    Pred[((size_t)t * B_ + (rbase + j)) * 64 + cidx] = acc[j] + bm;
}

// ---------------------------------------------------------------------------
// Kernel 3: fused totals GEMM + online column logsumexp + diagonal capture.
// 8 waves = 2 col-groups (32 cols) x 4 row-tiles; 2 accumulators/wave so one
// A fetch feeds two WMMAs (1.5 LDS loads per WMMA). Ping-pong LDS buffers with
// register prefetch + global_prefetch_b8 two tiles ahead; 1 barrier/iteration.
// ---------------------------------------------------------------------------
__global__ void k_nce(const float* __restrict__ Auxi, const int* __restrict__ tp,
                      const float* __restrict__ Pred, float* __restrict__ ncePart) {
  const int t  = blockIdx.x >> 6;
  const int cb = (blockIdx.x & 63) * 64;
  const int tid = threadIdx.x;
  const int toff = *tp + 1 + t;
  __shared__ __align__(16) float sEnc[2][64][68];   // ping-pong enc tiles
  __shared__ __align__(16) float sPred[64][68];
  __shared__ float sDiag[64];
  __shared__ float sM[8][32];
  __shared__ float sS[8][32];
  __shared__ float sRed[64];
  {   // stage Pred column tile once (L2-resident; reused for all 4096 rows)
    int r = tid >> 2;
    int cq = (tid & 3) * 16;
    const float* src = Pred + ((size_t)t * B_ + cb + r) * 64 + cq;
#pragma unroll
    for (int q = 0; q < 4; ++q)
      *(float4*)&sPred[r][cq + q * 4] = *(const float4*)(src + q * 4);
  }
  const int wave = tid >> 5, lane = tid & 31;
  const int colG = wave & 1;              // 32-col group
  const int rowQ = wave >> 1;             // 16-row tile within 64-row iter
  const int n = lane & 15, kb = (lane >> 4) * 2;
  const float* bRow0 = &sPred[colG * 32 + n][0];
  const float* bRow1 = &sPred[colG * 32 + 16 + n][0];
  const float* aBase = &sEnc[0][0][0] + (rowQ * 16 + n) * 68;
  const int myCol0 = cb + colG * 32;
  const int myCol1 = myCol0 + 16;
  const int jdiag = n - ((lane >> 4) << 3);   // valid in [0,8) for diag lanes
  float m0 = -INFINITY, s0 = 0.f, m1 = -INFINITY, s1 = 0.f;

  // staging: 64 rows x 64 floats per tile; 16 floats (4x float4) per thread
  const int sr = tid >> 2;
  const int sc = (tid & 3) * 16;
  float4 p0, p1, p2, p3;
  const float* src0 = Auxi + ((size_t)sr * S_ + toff) * 64 + sc;
  p0 = *(const float4*)(src0);
  p1 = *(const float4*)(src0 + 4);
  p2 = *(const float4*)(src0 + 8);
  p3 = *(const float4*)(src0 + 12);
  *(float4*)&sEnc[0][sr][sc]      = p0;
  *(float4*)&sEnc[0][sr][sc + 4]  = p1;
  *(float4*)&sEnc[0][sr][sc + 8]  = p2;
  *(float4*)&sEnc[0][sr][sc + 12] = p3;

  for (int it = 0; it < 64; ++it) {
    const int rb = it * 64;
    const int cur = it & 1;
    if (it + 1 < 64) {   // issue next tile's loads before computing this one
      const float* src = Auxi + ((size_t)(rb + 64 + sr) * S_ + toff) * 64 + sc;
      p0 = *(const float4*)(src);
      p1 = *(const float4*)(src + 4);
      p2 = *(const float4*)(src + 8);
      p3 = *(const float4*)(src + 12);
    }
    if (it + 2 < 64)     // warm L2/L0 two tiles ahead (global_prefetch_b8)
      __builtin_prefetch(Auxi + ((size_t)(rb + 128 + sr) * S_ + toff) * 64 + sc, 0, 3);
    __syncthreads();     // sEnc[cur] visible; prior reads of sEnc[cur^1] retired

    const float* aRow = aBase + cur * (64 * 68);
    v8f a0 = {0.f, 0.f, 0.f, 0.f, 0.f, 0.f, 0.f, 0.f};
    v8f a1 = {0.f, 0.f, 0.f, 0.f, 0.f, 0.f, 0.f, 0.f};
#pragma unroll
    for (int k0 = 0; k0 < 64; k0 += 4) {
      v2f av = *(const v2f*)(aRow + k0 + kb);
      v2f b0 = *(const v2f*)(bRow0 + k0 + kb);
      v2f b1 = *(const v2f*)(bRow1 + k0 + kb);
      a0 = __builtin_amdgcn_wmma_f32_16x16x4_f32(false, av, false, b0, (short)0, a0,
                                                 false, false);
      a1 = __builtin_amdgcn_wmma_f32_16x16x4_f32(false, av, false, b1, (short)0, a1,
                                                 false, false);
    }
    // diagonal capture (each column's diagonal row visited exactly once)
    const int rowBase = rb + rowQ * 16;
    if (rowBase == myCol0) {
      if (jdiag >= 0 && jdiag < 8) sDiag[colG * 32 + n] = a0[jdiag];
    }
    if (rowBase == myCol1) {
      if (jdiag >= 0 && jdiag < 8) sDiag[colG * 32 + 16 + n] = a1[jdiag];
    }
    // online logsumexp per column; lanes l and l^16 hold the same column
    {
      float tm = a0[0];
#pragma unroll
      for (int j = 1; j < 8; ++j) tm = fmaxf(tm, a0[j]);
      float tmax = fmaxf(tm, __shfl_xor(tm, 16, 32));
      float nm = fmaxf(m0, tmax);
      float corr = __expf(m0 - nm);
      float ls = 0.f;
#pragma unroll
      for (int j = 0; j < 8; ++j) ls += __expf(a0[j] - nm);
      ls += __shfl_xor(ls, 16, 32);
      s0 = s0 * corr + ls;
      m0 = nm;
    }
    {
      float tm = a1[0];
#pragma unroll
      for (int j = 1; j < 8; ++j) tm = fmaxf(tm, a1[j]);
      float tmax = fmaxf(tm, __shfl_xor(tm, 16, 32));
      float nm = fmaxf(m1, tmax);
      float corr = __expf(m1 - nm);
      float ls = 0.f;
#pragma unroll
      for (int j = 0; j < 8; ++j) ls += __expf(a1[j] - nm);
      ls += __shfl_xor(ls, 16, 32);
      s1 = s1 * corr + ls;
      m1 = nm;
    }
    if (it + 1 < 64) {   // fill the other buffer; its last readers passed the barrier
      const int nxt = cur ^ 1;
      *(float4*)&sEnc[nxt][sr][sc]      = p0;
      *(float4*)&sEnc[nxt][sr][sc + 4]  = p1;
      *(float4*)&sEnc[nxt][sr][sc + 8]  = p2;
      *(float4*)&sEnc[nxt][sr][sc + 12] = p3;
    }
  }
  if (lane < 16) {
    sM[wave][n] = m0;      sS[wave][n] = s0;
    sM[wave][16 + n] = m1; sS[wave][16 + n] = s1;
  }
  __syncthreads();
  if (tid < 64) {   // merge the 4 row-tile waves per column; form diag - lse
    const int cg = tid >> 5, cc = tid & 31;
    float mm = sM[cg][cc];
#pragma unroll
    for (int q = 1; q < 4; ++q) mm = fmaxf(mm, sM[cg + 2 * q][cc]);
    float ss = 0.f;
#pragma unroll
    for (int q = 0; q < 4; ++q) ss += sS[cg + 2 * q][cc] * __expf(sM[cg + 2 * q][cc] - mm);
    sRed[tid] = sDiag[tid] - (mm + logf(ss));
  }
  __syncthreads();
  for (int off = 32; off > 0; off >>= 1) {
    if (tid < off) sRed[tid] += sRed[tid + off];
    __syncthreads();
  }
  if (tid == 0) ncePart[blockIdx.x] = sRed[0];
}

// ---------------------------------------------------------------------------
// Head: fc1 -> deterministic batchnorm stats -> relu + fc2. Tiny (~16 MFLOP).
// ---------------------------------------------------------------------------
__global__ void k_fc1(const float* __restrict__ IM, const float* __restrict__ w1,
                      const float* __restrict__ b1, float* __restrict__ h) {
  int idx = blockIdx.x * 256 + threadIdx.x;
  int b = idx >> 5, o = idx & 31;
  const float* x = IM + (size_t)b * 64;
  const float* w = w1 + (size_t)o * 64;
  float acc = b1[o];
#pragma unroll
  for (int k = 0; k < 64; ++k) acc += x[k] * w[k];
  h[(size_t)b * 32 + o] = acc;
}

__global__ void k_bnstats(const float* __restrict__ h, float* __restrict__ stats) {
  int o = blockIdx.x;           // one block per channel: fixed order => deterministic
  int tid = threadIdx.x;
  float s = 0.f, s2 = 0.f;
  for (int i = tid; i < B_; i += 256) {
    float v = h[(size_t)i * 32 + o];
    s += v; s2 += v * v;
  }
  __shared__ float rs[256], rs2[256];
  rs[tid] = s; rs2[tid] = s2;
  __syncthreads();
  for (int off = 128; off > 0; off >>= 1) {
    if (tid < off) { rs[tid] += rs[tid + off]; rs2[tid] += rs2[tid + off]; }
    __syncthreads();
  }
  if (tid == 0) {
    float mu = rs[0] / (float)B_;
    float var = rs2[0] / (float)B_ - mu * mu;     // biased var (jnp.var default)
    stats[o] = mu;
    stats[32 + o] = rsqrtf(var + 1e-5f);
  }
}

__global__ void k_fc2(const float* __restrict__ h, const float* __restrict__ stats,
                      const float* __restrict__ g, const float* __restrict__ bb,
                      const float* __restrict__ w2, const float* __restrict__ b2,
                      float* __restrict__ out) {
  int idx = blockIdx.x * 256 + threadIdx.x;
  int b = idx >> 4, j = idx & 15;
  const float* hr = h + (size_t)b * 32;
  const float* w = w2 + (size_t)j * 32;
  float acc = b2[j];
#pragma unroll
  for (int o = 0; o < 32; ++o) {
    float v = (hr[o] - stats[o]) * stats[32 + o] * g[o] + bb[o];
    acc += fmaxf(v, 0.f) * w[o];
  }
  out[1 + (size_t)b * 16 + j] = acc;
}

__global__ void k_ncefinal(const float* __restrict__ part, float* __restrict__ out) {
  int tid = threadIdx.x;
  __shared__ float r[256];
  r[tid] = part[tid] + part[tid + 256];            // 512 partials, fixed order
  __syncthreads();
  for (int off = 128; off > 0; off >>= 1) {
    if (tid < off) r[tid] += r[tid + off];
    __syncthreads();
  }
  if (tid == 0) out[0] = r[0] * (-1.0f / (4096.0f * 8.0f));
}

// ---------------------------------------------------------------------------
extern "C" void kernel_launch(void* const* d_in, const int* in_sizes, int n_in,
                              void* d_out, int out_size, void* d_ws, size_t ws_size,
                              hipStream_t stream) {
  (void)in_sizes; (void)n_in; (void)out_size; (void)ws_size;
  const float* Main = (const float*)d_in[0];
  const float* Auxi = (const float*)d_in[1];
  const int*   tp   = (const int*)d_in[2];
  const float* Wp   = (const float*)d_in[3];
  const float* bp   = (const float*)d_in[4];
  const float* Wmap = (const float*)d_in[5];
  const float* bmap = (const float*)d_in[6];
  const float* w1   = (const float*)d_in[7];
  const float* b1   = (const float*)d_in[8];
  const float* bng  = (const float*)d_in[9];
  const float* bnb  = (const float*)d_in[10];
  const float* w2   = (const float*)d_in[11];
  const float* b2   = (const float*)d_in[12];
  float* out = (float*)d_out;
  float* ws  = (float*)d_ws;

  float* IM      = ws;                 // 4096*64   floats
  float* Pred    = ws + 262144;        // 8*4096*64 floats
  float* h       = ws + 2359296;       // 4096*32   floats
  float* ncePart = ws + 2490368;       // 512       floats
  float* stats   = ws + 2490880;       // 64        floats (mu | invstd)

  k_imcx    <<<4096, 256, 0, stream>>>(Main, tp, Wp, bp, IM);
  k_pred    <<<1024, 256, 0, stream>>>(IM, Wmap, bmap, Pred);
  k_nce     <<<512,  256, 0, stream>>>(Auxi, tp, Pred, ncePart);
  k_fc1     <<<512,  256, 0, stream>>>(IM, w1, b1, h);
  k_bnstats <<<32,   256, 0, stream>>>(h, stats);
  k_fc2     <<<256,  256, 0, stream>>>(h, stats, bng, bnb, w2, b2, out);
  k_ncefinal<<<1,    256, 0, stream>>>(ncePart, out);
}